// MedianFilter_60447369724077
// MI455X (gfx1250) — compile-verified
//
#include <hip/hip_runtime.h>

// ---------------------------------------------------------------------------
// 5x5 median blur, zero padding, fp32, 16x3x512x512.  VALU-bound.
// Algorithm: sorted-columns (amortized across 4 outputs/thread) -> sorted
// rank-rows -> doubly-sorted matrix -> 13-candidate set -> forgetful
// selection -> v_med3_f32.  Staging: global_load_async_to_lds_b32 (ASYNCcnt).
// ---------------------------------------------------------------------------

#define S2(a, b)                         \
    do {                                 \
        float _t_ = fminf((a), (b));     \
        (b) = fmaxf((a), (b));           \
        (a) = _t_;                       \
    } while (0)

// 5-CE sort of 4 (classic network, provably correct).
__device__ __forceinline__ void sort4(float& a, float& b, float& c, float& d) {
    S2(a, b); S2(c, d); S2(a, c); S2(b, d); S2(b, c);
}

// Insert e into sorted a<=b<=c<=d (one bubble pass, 4 CE): a<=b<=c<=d<=e.
__device__ __forceinline__ void insert5(float& a, float& b, float& c, float& d, float& e) {
    S2(d, e); S2(c, d); S2(b, c); S2(a, b);
}

// 9-CE full sort of 5 built from verified primitives.
__device__ __forceinline__ void sort5(float& a, float& b, float& c, float& d, float& e) {
    sort4(a, b, c, d);
    insert5(a, b, c, d, e);
}

// Compare-exchange network: global min -> v[0], global max -> v[S-1],
// multiset preserved. Cost ceil(3S/2)-2 exchanges.
template <int S>
__device__ __forceinline__ void minmax_ends(float* v) {
#pragma unroll
    for (int i = 0; i + 1 < S; i += 2) S2(v[i], v[i + 1]);
#pragma unroll
    for (int i = 2; i < S; i += 2) S2(v[0], v[i]);
#pragma unroll
    for (int i = 1; i + 2 < S; i += 2) S2(v[i], v[i + 2]);
    if constexpr ((S & 1) != 0) S2(v[S - 2], v[S - 1]);
}

// Forgetful step: drop current min & max, append next element (size S -> S-1).
template <int S>
__device__ __forceinline__ void forget_step(float* v, float nw) {
    minmax_ends<S>(v);
#pragma unroll
    for (int i = 0; i + 2 < S; ++i) v[i] = v[i + 1];
    v[S - 2] = nw;
}

// Exact median (rank 7) of 13 values: forgetful selection from 8 candidates.
// At each step: (s-1) + prior-max-discards = 7 >= needed witnesses. Ends with
// 3 values bracketing the median -> single v_med3_f32.
__device__ __forceinline__ float median13(const float c[13]) {
    float v[8];
#pragma unroll
    for (int i = 0; i < 8; ++i) v[i] = c[i];
    forget_step<8>(v, c[8]);
    forget_step<7>(v, c[9]);
    forget_step<6>(v, c[10]);
    forget_step<5>(v, c[11]);
    forget_step<4>(v, c[12]);
    return __builtin_amdgcn_fmed3f(v[0], v[1], v[2]);
}

constexpr int IMG_W  = 512;
constexpr int IMG_H  = 512;
constexpr int PLANES = 16 * 3;
constexpr int BX     = 16;           // threads x
constexpr int BY     = 16;           // threads y
constexpr int OPT    = 4;            // outputs per thread (horizontal)
constexpr int TX     = BX * OPT;     // 64: output tile width per block
constexpr int TY     = BY;           // 16: output tile height per block
constexpr int LW     = TX + 5;       // 69: LDS stride (68 halo cols + 1 pad -> bank-conflict-free)
constexpr int LH     = TY + 4;       // 20: LDS tile height incl. halo
constexpr int NCELL  = LW * LH;      // 1380 cells
constexpr int NTHR   = BX * BY;      // 256 threads = 8 wave32

__global__ __launch_bounds__(NTHR) void MedianFilter5x5_kernel(
    const float* __restrict__ img, float* __restrict__ out) {
    __shared__ float tile[NCELL];

    const int plane = blockIdx.z;
    const float* __restrict__ src = img + (size_t)plane * (IMG_W * IMG_H);

    const int x0  = blockIdx.x * TX - 2;   // global x of tile col 0
    const int y0  = blockIdx.y * TY - 2;   // global y of tile row 0
    const int lx  = threadIdx.x;
    const int ly  = threadIdx.y;
    const int tid = ly * BX + lx;

    // ---- Stage halo tile into LDS: async DMA for in-range, 0 for padding --
    for (int base = 0; base < NCELL; base += NTHR) {
        const int cell = base + tid;
        if (cell < NCELL) {
            const int r  = cell / LW;
            const int c  = cell - r * LW;
            const int gy = y0 + r;
            const int gx = x0 + c;
            if ((unsigned)gx < (unsigned)IMG_W && (unsigned)gy < (unsigned)IMG_H) {
                const float* gptr   = src + gy * IMG_W + gx;
                unsigned     ldsoff = (unsigned)(size_t)(void*)&tile[cell];
                asm volatile("global_load_async_to_lds_b32 %0, %1, off"
                             :
                             : "v"(ldsoff), "v"(gptr)
                             : "memory");
            } else {
                tile[cell] = 0.0f;  // zero padding
            }
        }
    }
    asm volatile("s_wait_asynccnt 0x0" ::: "memory");
    __syncthreads();

    // ---- Load & sort the 8 shared window columns (5 deep) -----------------
    const int lx4 = lx * OPT;
    float col[8][5];
#pragma unroll
    for (int c8 = 0; c8 < 8; ++c8) {
#pragma unroll
        for (int r = 0; r < 5; ++r)
            col[c8][r] = tile[(ly + r) * LW + (lx4 + c8)];
        sort5(col[c8][0], col[c8][1], col[c8][2], col[c8][3], col[c8][4]);
    }

    // ---- 4 outputs: row-sort the 5 rank-rows, pick 13 candidates ----------
    float res[OPT];
#pragma unroll
    for (int d = 0; d < OPT; ++d) {
        float cand[13];
        {   // row 0 (column minima): keep sorted positions 3,4
            float m0 = col[d][0], m1 = col[d+1][0], m2 = col[d+2][0], m3 = col[d+3][0], m4 = col[d+4][0];
            sort5(m0, m1, m2, m3, m4);
            cand[0] = m3; cand[1] = m4;
        }
        {   // row 1: positions 2,3,4
            float m0 = col[d][1], m1 = col[d+1][1], m2 = col[d+2][1], m3 = col[d+3][1], m4 = col[d+4][1];
            sort5(m0, m1, m2, m3, m4);
            cand[2] = m2; cand[3] = m3; cand[4] = m4;
        }
        {   // row 2: positions 1,2,3
            float m0 = col[d][2], m1 = col[d+1][2], m2 = col[d+2][2], m3 = col[d+3][2], m4 = col[d+4][2];
            sort5(m0, m1, m2, m3, m4);
            cand[5] = m1; cand[6] = m2; cand[7] = m3;
        }
        {   // row 3: positions 0,1,2
            float m0 = col[d][3], m1 = col[d+1][3], m2 = col[d+2][3], m3 = col[d+3][3], m4 = col[d+4][3];
            sort5(m0, m1, m2, m3, m4);
            cand[8] = m0; cand[9] = m1; cand[10] = m2;
        }
        {   // row 4 (column maxima): positions 0,1
            float m0 = col[d][4], m1 = col[d+1][4], m2 = col[d+2][4], m3 = col[d+3][4], m4 = col[d+4][4];
            sort5(m0, m1, m2, m3, m4);
            cand[11] = m0; cand[12] = m1;
        }
        res[d] = median13(cand);
    }

    // ---- Store 4 consecutive outputs as one b128 --------------------------
    const int gx_out = blockIdx.x * TX + lx4;
    const int gy_out = blockIdx.y * TY + ly;
    float4 o;
    o.x = res[0]; o.y = res[1]; o.z = res[2]; o.w = res[3];
    *(float4*)(out + (size_t)plane * (IMG_W * IMG_H) + (size_t)gy_out * IMG_W + gx_out) = o;
}

extern "C" void kernel_launch(void* const* d_in, const int* in_sizes, int n_in,
                              void* d_out, int out_size, void* d_ws, size_t ws_size,
                              hipStream_t stream) {
    (void)in_sizes; (void)n_in; (void)d_ws; (void)ws_size; (void)out_size;
    const float* img = (const float*)d_in[0];  // (16,3,512,512) fp32
    // d_in[1] is kernel_size == 5 (kernel is specialized for k=5)
    float* out = (float*)d_out;

    dim3 block(BX, BY, 1);
    dim3 grid(IMG_W / TX, IMG_H / TY, PLANES);
    MedianFilter5x5_kernel<<<grid, block, 0, stream>>>(img, out);
}